// GCNLayer_197568495782
// MI455X (gfx1250) — compile-verified
//
#include <hip/hip_runtime.h>

#define NUM_USERS 100000
#define NUM_ITEMS 50000
#define NUM_EDGES 1250000
#define EMB 64

typedef __attribute__((ext_vector_type(2))) float v2f;
typedef __attribute__((ext_vector_type(8))) float v8f;

// ---------------- K0: zero degree accumulators ----------------
__global__ void zero_deg_kernel(float* __restrict__ deg_u, float* __restrict__ deg_i) {
    int t = blockIdx.x * blockDim.x + threadIdx.x;
    if (t < NUM_USERS) deg_u[t] = 0.0f;
    if (t < NUM_ITEMS) deg_i[t] = 0.0f;
}

// ---------------- K1: msg = emb (the "+ u_emb" / "+ i_emb" bias), float4 ----------------
__global__ void init_msg_kernel(const float4* __restrict__ u_emb, const float4* __restrict__ i_emb,
                                float4* __restrict__ u_msg, float4* __restrict__ i_msg) {
    long t = (long)blockIdx.x * blockDim.x + threadIdx.x;
    const long NU = (long)NUM_USERS * EMB / 4;  // 1.6M float4
    const long NI = (long)NUM_ITEMS * EMB / 4;  // 0.8M float4
    if (t < NU) {
        u_msg[t] = u_emb[t];
    } else if (t < NU + NI) {
        i_msg[t - NU] = i_emb[t - NU];
    }
}

// ---------------- K2: degree counts via L2 atomics ----------------
__global__ void degree_kernel(const int* __restrict__ ui, const int* __restrict__ ii,
                              float* __restrict__ deg_u, float* __restrict__ deg_i) {
    int e = blockIdx.x * blockDim.x + threadIdx.x;
    if (e < NUM_EDGES) {
        atomicAdd(&deg_u[ui[e]], 1.0f);
        atomicAdd(&deg_i[ii[e]], 1.0f);
    }
}

// ---------------- K3: bidirectional weighted scatter; 64 lanes per edge ----------------
__global__ void scatter_kernel(const float* __restrict__ u_emb, const float* __restrict__ i_emb,
                               const float* __restrict__ w,
                               const int* __restrict__ ui, const int* __restrict__ ii,
                               const float* __restrict__ deg_u, const float* __restrict__ deg_i,
                               float* __restrict__ u_msg, float* __restrict__ i_msg) {
    long t = (long)blockIdx.x * blockDim.x + threadIdx.x;
    long e = t >> 6;          // edge id
    int  d = (int)(t & 63);   // embedding dim
    if (e >= NUM_EDGES) return;

    int u = ui[e];
    int i = ii[e];
    float du = fmaxf(deg_u[u], 1.0f);
    float di = fmaxf(deg_i[i], 1.0f);
    float ew = w[e] / sqrtf(du * di);

    // coalesced 64-float runs per edge; f32 adds resolve in the 192MB L2
    atomicAdd(&u_msg[(long)u * EMB + d], i_emb[(long)i * EMB + d] * ew);
    atomicAdd(&i_msg[(long)i * EMB + d], u_emb[(long)u * EMB + d] * ew);
}

// ---------------- K4: out = relu(msg @ W^T), f32 WMMA 16x16x4 chained over K=64 ----------------
// One wave (32 lanes) computes one 16x16 output tile.
// A tile  (16x4 f32):  lanes 0-15 -> M=lane, {K, K+1}; lanes 16-31 -> M=lane-16, {K+2, K+3}
// B tile  (4x16 f32):  B[k][n] = W[n][k]; lanes 0-15 -> N=lane, {K, K+1}; lanes 16-31 -> N=lane-16, {K+2,K+3}
// C/D     (16x16 f32): VGPR v: lanes 0-15 -> M=v, N=lane; lanes 16-31 -> M=v+8, N=lane-16
__global__ __launch_bounds__(32) void gemm_relu_kernel(const float* __restrict__ msg,
                                                       const float* __restrict__ W,
                                                       float* __restrict__ out) {
    const int rowTile = blockIdx.x;   // 16 rows of msg
    const int colTile = blockIdx.y;   // 16 output columns
    const int lane = threadIdx.x;
    const int half = lane >> 4;       // 0 or 1
    const int l15  = lane & 15;

    const float* Abase = msg + ((long)rowTile * 16 + l15) * EMB + half * 2;
    const float* Bbase = W   + ((long)colTile * 16 + l15) * EMB + half * 2;

    v8f c = {0.f, 0.f, 0.f, 0.f, 0.f, 0.f, 0.f, 0.f};
#pragma unroll
    for (int kk = 0; kk < EMB; kk += 4) {
        v2f a = *(const v2f*)(Abase + kk);   // 8B-aligned (kk even, half*2 even)
        v2f b = *(const v2f*)(Bbase + kk);
        // 8 args: (neg_a, A, neg_b, B, c_mod, C, reuse_a, reuse_b)
        c = __builtin_amdgcn_wmma_f32_16x16x4_f32(false, a, false, b, (short)0, c, false, false);
    }

    const long outBase = (long)rowTile * 16 * EMB + colTile * 16;
#pragma unroll
    for (int v = 0; v < 8; ++v) {
        const int m = v + half * 8;
        const float x = c[v];
        out[outBase + (long)m * EMB + l15] = x > 0.f ? x : 0.f;
    }
}

// ---------------- launch ----------------
extern "C" void kernel_launch(void* const* d_in, const int* in_sizes, int n_in,
                              void* d_out, int out_size, void* d_ws, size_t ws_size,
                              hipStream_t stream) {
    const float* u_emb = (const float*)d_in[0];
    const float* i_emb = (const float*)d_in[1];
    const float* w     = (const float*)d_in[2];
    const int*   ui    = (const int*)d_in[3];
    const int*   ii    = (const int*)d_in[4];
    const float* W_u   = (const float*)d_in[5];
    const float* W_i   = (const float*)d_in[6];

    float* out   = (float*)d_out;
    float* new_u = out;                         // 100000 x 64
    float* new_i = out + (long)NUM_USERS * EMB; // 50000 x 64

    float* ws    = (float*)d_ws;
    float* deg_u = ws;
    float* deg_i = deg_u + NUM_USERS;
    float* u_msg = deg_i + NUM_ITEMS;
    float* i_msg = u_msg + (long)NUM_USERS * EMB;

    // K0: zero degrees
    zero_deg_kernel<<<(NUM_USERS + 255) / 256, 256, 0, stream>>>(deg_u, deg_i);

    // K1: msg accumulators start at the embeddings
    long nvec = ((long)NUM_USERS + NUM_ITEMS) * EMB / 4;
    init_msg_kernel<<<(int)((nvec + 255) / 256), 256, 0, stream>>>(
        (const float4*)u_emb, (const float4*)i_emb, (float4*)u_msg, (float4*)i_msg);

    // K2: degrees
    degree_kernel<<<(NUM_EDGES + 255) / 256, 256, 0, stream>>>(ui, ii, deg_u, deg_i);

    // K3: bidirectional scatter (64 lanes per edge)
    long nthreads = (long)NUM_EDGES * EMB;
    scatter_kernel<<<(int)((nthreads + 255) / 256), 256, 0, stream>>>(
        u_emb, i_emb, w, ui, ii, deg_u, deg_i, u_msg, i_msg);

    // K4: fused GEMM + ReLU via f32 WMMA
    dim3 gu(NUM_USERS / 16, EMB / 16);  // 6250 x 4 tiles
    gemm_relu_kernel<<<gu, 32, 0, stream>>>(u_msg, W_u, new_u);
    dim3 gi(NUM_ITEMS / 16, EMB / 16);  // 3125 x 4 tiles
    gemm_relu_kernel<<<gi, 32, 0, stream>>>(i_msg, W_i, new_i);
}